// GaussianMixture_45079976738961
// MI455X (gfx1250) — compile-verified
//
#include <hip/hip_runtime.h>
#include <math.h>

// CDNA5 / gfx1250, wave32. FP32 WMMA path: V_WMMA_F32_16X16X4_F32.
typedef __attribute__((ext_vector_type(2))) float v2f;
typedef __attribute__((ext_vector_type(8))) float v8f;

#define MAX_M 512   // reference M; coef table = M*8 floats = 16KB LDS

// ---------------------------------------------------------------------------
// Prep kernel: one block. Builds per-component coefficient rows c_j[0..7] s.t.
//   logit_ij = f_i . c_j  with f_i = [x0^2, x0x1, x1x0, x1^2, x0, x1, 1, 0].
// Includes the log_softmax(w) reduction (max + sum-exp over M).
// ---------------------------------------------------------------------------
__global__ __launch_bounds__(MAX_M)
void gmm_prep_kernel(const float* __restrict__ mu,
                     const float* __restrict__ Amat,
                     const float* __restrict__ w,
                     float* __restrict__ coef, int M) {
    __shared__ float red[MAX_M];
    const int j = threadIdx.x;
    const float wj = (j < M) ? w[j] : -INFINITY;

    // logsumexp over w (block tree reduction)
    red[j] = wj;
    __syncthreads();
    for (int off = MAX_M / 2; off > 0; off >>= 1) {
        if (j < off) red[j] = fmaxf(red[j], red[j + off]);
        __syncthreads();
    }
    const float wmax = red[0];
    __syncthreads();
    red[j] = (j < M) ? __expf(wj - wmax) : 0.0f;
    __syncthreads();
    for (int off = MAX_M / 2; off > 0; off >>= 1) {
        if (j < off) red[j] += red[j + off];
        __syncthreads();
    }
    const float lse_w = wmax + __logf(red[0]);

    if (j < M) {
        const float a00 = Amat[j * 4 + 0], a01 = Amat[j * 4 + 1];
        const float a10 = Amat[j * 4 + 2], a11 = Amat[j * 4 + 3];
        // G = (A A^T)/2  (symmetric PSD)
        const float G00 = 0.5f * (a00 * a00 + a01 * a01);
        const float G01 = 0.5f * (a00 * a10 + a01 * a11);
        const float G10 = G01;
        const float G11 = 0.5f * (a10 * a10 + a11 * a11);
        const float det = G00 * G11 - G01 * G10;

        const float m0 = mu[j * 2 + 0], m1 = mu[j * 2 + 1];
        const float Gm0 = G00 * m0 + G01 * m1;
        const float Gm1 = G10 * m0 + G11 * m1;
        const float mumu = m0 * Gm0 + m1 * Gm1;
        const float logw = (wj - lse_w) + 0.5f * __logf(det);

        float* c = coef + j * 8;
        c[0] = -G00; c[1] = -G01; c[2] = -G10; c[3] = -G11;
        c[4] = 2.0f * Gm0; c[5] = 2.0f * Gm1;
        c[6] = logw - mumu;
        c[7] = 0.0f;
    }
}

// ---------------------------------------------------------------------------
// Main kernel: each wave32 owns 16 points. B (point features) is loop-
// invariant in 4 VGPRs; loop over M/16 component tiles, 2x WMMA f32 16x16x4
// per tile (K=8), online logsumexp over the 8 f32 accumulators per lane.
// Lane l and lane l^16 hold disjoint component halves of point (l&15);
// merged at the end with one shfl_xor.
// ---------------------------------------------------------------------------
__global__ __launch_bounds__(256)
void gmm_wmma_logsumexp_kernel(const float* __restrict__ sample,
                               const float* __restrict__ coefG,
                               float* __restrict__ out, int N, int M) {
    __shared__ float coefLds[MAX_M * 8];   // 16 KB of 320 KB/WGP

    const int tid = threadIdx.x;
    // Stage coefficient table into LDS (hot for all 32 tiles of the sweep).
    for (int i = tid; i < M * 8; i += 256) coefLds[i] = coefG[i];
    __syncthreads();

    const int lane = tid & 31;
    const bool hi  = lane >= 16;          // upper half-wave (K/comp offset)
    const int wave = tid >> 5;
    const int base = blockIdx.x * 128 + wave * 16;   // 8 waves * 16 pts
    const int prow = lane & 15;
    int p = base + prow;
    const int pc = (p < N) ? p : (N - 1);            // clamp for safe load

    const float x0 = sample[2 * pc + 0];
    const float x1 = sample[2 * pc + 1];

    // B matrix (4x16, K x points), per ISA layout:
    //  VGPR v: lanes 0-15 -> K=v, lanes 16-31 -> K=v+2; column N = lane&15.
    v2f B0, B1;
    if (!hi) { B0.x = x0 * x0; B0.y = x0 * x1; B1.x = x0;   B1.y = x1;   }
    else     { B0.x = x0 * x1; B0.y = x1 * x1; B1.x = 1.0f; B1.y = 0.0f; }

    float m = -INFINITY;   // running max of logits over this lane's comps
    float s = 0.0f;        // running rescaled sum of exp

    const int tiles = M >> 4;            // M multiple of 16 (M=512)
    const int kbase = hi ? 2 : 0;        // A: lanes>=16 hold K=2,3 of each row

    for (int t = 0; t < tiles; ++t) {
        const int c = (t << 4) + prow;   // A row = component (lane&15)
        const float* cr = coefLds + c * 8 + kbase;
        v2f A0, A1;                      // ds_load_b64 pairs
        A0.x = cr[0]; A0.y = cr[1];      // coefs k = kbase, kbase+1
        A1.x = cr[4]; A1.y = cr[5];      // coefs k = kbase+4, kbase+5

        v8f acc = {};
        acc = __builtin_amdgcn_wmma_f32_16x16x4_f32(
                  false, A0, false, B0, (short)0, acc, false, false);
        acc = __builtin_amdgcn_wmma_f32_16x16x4_f32(
                  false, A1, false, B1, (short)0, acc, false, false);
        // acc[v] = logit of comp (t*16 + v + (hi?8:0)) for point (lane&15)

        float tmax = acc[0];
        #pragma unroll
        for (int v = 1; v < 8; ++v) tmax = fmaxf(tmax, acc[v]);
        const float nm = fmaxf(m, tmax);
        float ssum = 0.0f;
        #pragma unroll
        for (int v = 0; v < 8; ++v) ssum += __expf(acc[v] - nm);
        s = s * __expf(m - nm) + ssum;   // exp(-inf)=0 handles first iter
        m = nm;
    }

    // Merge the two half-wave partials for each point (disjoint comp sets).
    const float pm = __shfl_xor(m, 16, 32);
    const float ps = __shfl_xor(s, 16, 32);
    const float fm = fmaxf(m, pm);
    const float fs = s * __expf(m - fm) + ps * __expf(pm - fm);

    if (!hi && p < N) out[p] = fm + __logf(fs);
}

extern "C" void kernel_launch(void* const* d_in, const int* in_sizes, int n_in,
                              void* d_out, int out_size, void* d_ws, size_t ws_size,
                              hipStream_t stream) {
    const float* sample = (const float*)d_in[0];   // (N,2)
    const float* mu     = (const float*)d_in[1];   // (M,2)
    const float* Amat   = (const float*)d_in[2];   // (M,2,2)
    const float* w      = (const float*)d_in[3];   // (M,1)
    float* out = (float*)d_out;                    // (N,1)

    const int N = in_sizes[0] / 2;
    const int M = in_sizes[3];                     // w has M elements

    float* coef = (float*)d_ws;                    // M*8 floats = 16 KB

    gmm_prep_kernel<<<1, MAX_M, 0, stream>>>(mu, Amat, w, coef, M);

    const int blocks = (N + 127) / 128;            // 128 points per 256-thr block
    gmm_wmma_logsumexp_kernel<<<blocks, 256, 0, stream>>>(sample, coef, out, N, M);
}